// Lift3DLocalFusionGraspNet_16733192585552
// MI455X (gfx1250) — compile-verified
//
#include <hip/hip_runtime.h>

typedef __attribute__((ext_vector_type(2))) float v2f;
typedef __attribute__((ext_vector_type(8))) float v8f;

#define BB   4
#define SS   4096
#define NC   8192
#define CC   512     // ref_feat channels
#define CSD  256     // seed channels / output channels

#define NCHUNK 8
#define CHUNK  (NC / NCHUNK)   // 1024 refs per chunk

// ---------------------------------------------------------------------------
// Full-precision f32 WMMA: D(16x16) = A(16x4) * B(4x16) + C
// A layout: lane m=lane%16 holds row M=m; VGPR0 = K=(half?2:0), VGPR1 = K=(half?3:1)
// B layout: VGPR0 = row K=(half?2:0) cols N=lane%16; VGPR1 = K=(half?3:1)
// C/D layout: VGPR v, lane -> M = v + 8*(lane/16), N = lane%16
// ---------------------------------------------------------------------------
__device__ __forceinline__ v8f wmma_f32x4(v2f a, v2f b, v8f c) {
    return __builtin_amdgcn_wmma_f32_16x16x4_f32(
        /*neg_a=*/false, a, /*neg_b=*/false, b,
        /*c_mod=*/(short)0, c, /*reuse_a=*/false, /*reuse_b=*/false);
}

// ---------------------------------------------------------------------------
// Kernel 0: r2[b,n] = |ref_xyz[b,n]|^2  (one-time, 32K values)
// ---------------------------------------------------------------------------
__global__ __launch_bounds__(256) void r2_kernel(
    const float* __restrict__ rxyz, float* __restrict__ r2arr)
{
    const int i = blockIdx.x * blockDim.x + threadIdx.x;   // B*NC
    const float* p = rxyz + (size_t)i * 3;
    r2arr[i] = p[0] * p[0] + p[1] * p[1] + p[2] * p[2];
}

// ---------------------------------------------------------------------------
// Kernel 1: WMMA nearest-neighbor.  One wave per (16-query block, Nc chunk).
// qr tile (16x16) from one V_WMMA_F32_16X16X4_F32 (K = x,y,z,0); epilogue
// d2 = q2 + r2 - 2*qr with running (best,idx) per accumulator slot.
// 4*256*8 = 8192 waves -> machine stays full (vs 512 waves thread-per-query).
// ---------------------------------------------------------------------------
__global__ __launch_bounds__(256) void nn_wmma_kernel(
    const float* __restrict__ qxyz,    // (B,S,3)
    const float* __restrict__ rxyz,    // (B,Nc,3)
    const float* __restrict__ r2arr,   // (B,Nc)
    float* __restrict__ pbest,         // (B*S, NCHUNK)
    int*   __restrict__ pidx)          // (B*S, NCHUNK)
{
    const int wave = blockIdx.x * 8 + (threadIdx.x >> 5);
    const int lane = threadIdx.x & 31;
    const int m    = lane & 15;
    const int half = lane >> 4;

    const int b   = wave >> 11;        // 256*8 waves per batch
    const int rem = wave & 2047;
    const int q0  = (rem >> 3) * 16;   // query block
    const int nb  = (rem & 7) * CHUNK; // chunk start

    // A fragment: 16 queries x K=(x,y,z,0)
    const float* qp = qxyz + (size_t)(b * SS + q0 + m) * 3;
    v2f a;
    a.x = qp[2 * half];                  // half0: x, half1: z
    a.y = half ? 0.0f : qp[1];           // half0: y, half1: 0 (K=3 pad)

    // q2 per D slot: slot v holds query M = v + 8*half
    float q2v[8];
    #pragma unroll
    for (int v = 0; v < 8; ++v) {
        const float* qq = qxyz + (size_t)(b * SS + q0 + v + 8 * half) * 3;
        q2v[v] = qq[0] * qq[0] + qq[1] * qq[1] + qq[2] * qq[2];
    }

    float best[8];
    int   bidx[8];
    #pragma unroll
    for (int v = 0; v < 8; ++v) { best[v] = 3.4e38f; bidx[v] = 0; }

    const float* rb  = rxyz  + (size_t)b * NC * 3;
    const float* r2b = r2arr + (size_t)b * NC;

    for (int n0 = nb; n0 < nb + CHUNK; n0 += 16) {
        const float* rp = rb + (size_t)(n0 + m) * 3;
        __builtin_prefetch(rp + 48, 0, 0);          // next tile -> global_prefetch_b8
        v2f bf;
        bf.x = rp[2 * half];                        // half0: x, half1: z
        bf.y = half ? 0.0f : rp[1];                 // half0: y, half1: 0
        const float r2 = r2b[n0 + m];

        v8f zero = {};
        v8f qr = wmma_f32x4(a, bf, zero);           // 256 dot products

        const int cand = n0 + m;                    // column N = m
        #pragma unroll
        for (int v = 0; v < 8; ++v) {
            const float d2 = __builtin_fmaf(-2.0f, qr[v], q2v[v] + r2);
            const bool lt = d2 < best[v];
            best[v] = lt ? d2 : best[v];
            bidx[v] = lt ? cand : bidx[v];
        }
    }

    // argmin across the 16 lanes of this half (candidates for each query row);
    // tie-break to lower index == first-occurrence argmin.
    #pragma unroll
    for (int v = 0; v < 8; ++v) {
        float bv = best[v];
        int   iv = bidx[v];
        #pragma unroll
        for (int off = 8; off >= 1; off >>= 1) {
            const float ob = __shfl_xor(bv, off, 32);
            const int   oi = __shfl_xor(iv, off, 32);
            if (ob < bv || (ob == bv && oi < iv)) { bv = ob; iv = oi; }
        }
        if (m == 0) {
            const int q = b * SS + q0 + v + 8 * half;
            pbest[(size_t)q * NCHUNK + (rem & 7)] = bv;
            pidx [(size_t)q * NCHUNK + (rem & 7)] = iv;
        }
    }
}

// ---------------------------------------------------------------------------
// Kernel 2: reduce the NCHUNK partials per query (ascending chunk order +
// strict < keeps the earliest minimum).
// ---------------------------------------------------------------------------
__global__ __launch_bounds__(256) void nn_reduce_kernel(
    const float* __restrict__ pbest, const int* __restrict__ pidx,
    int* __restrict__ idx)
{
    const int q = blockIdx.x * blockDim.x + threadIdx.x;  // B*S
    float best = 3.4e38f;
    int   bi   = 0;
    #pragma unroll
    for (int c = 0; c < NCHUNK; ++c) {
        const float v = pbest[(size_t)q * NCHUNK + c];
        const int   i = pidx [(size_t)q * NCHUNK + c];
        if (v < best) { best = v; bi = i; }
    }
    idx[q] = bi;
}

// ---------------------------------------------------------------------------
// Kernel 3: Weff = Wc2 @ W_seed   (256 x 512, K = 256), one wave per 16x16 tile
// ---------------------------------------------------------------------------
__global__ __launch_bounds__(256) void weff_kernel(
    const float* __restrict__ Wconcat,  // (256,512)
    const float* __restrict__ Wseed,    // (256,512)
    float* __restrict__ Weff)           // (256,512)
{
    const int wave = (blockIdx.x * blockDim.x + threadIdx.x) >> 5;
    const int lane = threadIdx.x & 31;
    const int m    = lane & 15;
    const int half = lane >> 4;

    const int o0 = (wave >> 5) * 16;    // 16 o-tiles
    const int c0 = (wave & 31) * 16;    // 32 c-tiles

    v8f acc = {};
    #pragma unroll 8
    for (int k0 = 0; k0 < CSD; k0 += 4) {
        const int ka = k0 + 2 * half;
        v2f a = *(const v2f*)&Wconcat[(size_t)(o0 + m) * 512 + 256 + ka];
        v2f bf;
        bf.x = Wseed[(size_t)(ka + 0) * 512 + c0 + m];
        bf.y = Wseed[(size_t)(ka + 1) * 512 + c0 + m];
        acc = wmma_f32x4(a, bf, acc);
    }
    #pragma unroll
    for (int v = 0; v < 8; ++v)
        Weff[(size_t)(o0 + v + 8 * half) * 512 + c0 + m] = acc[v];
}

// ---------------------------------------------------------------------------
// Kernel 4: beff[o] = b_concat[o] + sum_j Wc2[o][j] * b_seed[j]
// ---------------------------------------------------------------------------
__global__ __launch_bounds__(256) void beff_kernel(
    const float* __restrict__ Wconcat,
    const float* __restrict__ bseed,
    const float* __restrict__ bconcat,
    float* __restrict__ beff)
{
    const int o = threadIdx.x;
    float acc = bconcat[o];
    #pragma unroll 8
    for (int j = 0; j < CSD; ++j)
        acc += Wconcat[(size_t)o * 512 + 256 + j] * bseed[j];
    beff[o] = acc;
}

// ---------------------------------------------------------------------------
// Kernel 5: fused output GEMM with in-K gather (ref_feat is L2-resident).
//   out[b,o,s] = sum_k Wc1[o,k]*seed[b,k,s]
//              + sum_c Weff[o,c]*ref_feat[b,c,idx[b,s]] + beff[o]
// ---------------------------------------------------------------------------
__global__ __launch_bounds__(256) void fused_out_kernel(
    const float* __restrict__ Wconcat,  // (256,512): Wc1 = cols 0..255
    const float* __restrict__ Weff,     // (256,512)
    const float* __restrict__ seed,     // (B,256,S)
    const float* __restrict__ reff,     // (B,512,Nc)
    const int*   __restrict__ idx,      // (B,S)
    const float* __restrict__ beff,     // (256)
    float*       __restrict__ out)      // (B,256,S)
{
    const int wave = blockIdx.x * 8 + (threadIdx.x >> 5);
    const int lane = threadIdx.x & 31;
    const int m    = lane & 15;
    const int half = lane >> 4;

    const int b   = wave >> 12;              // 16*256 tiles per batch
    const int rem = wave & 4095;
    const int o0  = (rem >> 8) * 16;         // 16 o-tiles
    const int s0  = (rem & 255) * 16;        // 256 s-tiles

    const int myIdx = idx[b * SS + s0 + m];  // gather column for N = m

    const float* seedB = seed + (size_t)b * CSD * SS;
    const float* reffB = reff + (size_t)b * CC * NC;

    v8f acc = {};

    // ---- phase 1: K over 256 seed channels with Wc1 ----
    #pragma unroll 4
    for (int k0 = 0; k0 < CSD; k0 += 4) {
        const int ka = k0 + 2 * half;
        v2f a = *(const v2f*)&Wconcat[(size_t)(o0 + m) * 512 + ka];
        v2f bf;
        bf.x = seedB[(size_t)(ka + 0) * SS + s0 + m];
        bf.y = seedB[(size_t)(ka + 1) * SS + s0 + m];
        acc = wmma_f32x4(a, bf, acc);
    }

    // ---- phase 2: K over 512 gathered ref_feat channels with Weff ----
    #pragma unroll 4
    for (int k0 = 0; k0 < CC; k0 += 4) {
        const int ka = k0 + 2 * half;
        v2f a = *(const v2f*)&Weff[(size_t)(o0 + m) * 512 + ka];
        v2f bf;
        const float* g = &reffB[(size_t)ka * NC + myIdx];
        __builtin_prefetch(g + 4 * NC, 0, 0);   // next k-step gather line
        bf.x = g[0];
        bf.y = g[NC];
        acc = wmma_f32x4(a, bf, acc);
    }

    // ---- bias + store: D lane/VGPR -> (M = v + 8*half, N = m) ----
    float* outB = out + (size_t)b * CSD * SS;
    #pragma unroll
    for (int v = 0; v < 8; ++v) {
        const int o = o0 + v + 8 * half;
        outB[(size_t)o * SS + s0 + m] = acc[v] + beff[o];
    }
}

// ---------------------------------------------------------------------------
extern "C" void kernel_launch(void* const* d_in, const int* in_sizes, int n_in,
                              void* d_out, int out_size, void* d_ws, size_t ws_size,
                              hipStream_t stream) {
    const float* qxyz    = (const float*)d_in[0];
    const float* rxyz    = (const float*)d_in[1];
    const float* reff    = (const float*)d_in[2];
    const float* seed    = (const float*)d_in[3];
    const float* Wseed   = (const float*)d_in[4];
    const float* bseed   = (const float*)d_in[5];
    const float* Wconcat = (const float*)d_in[6];
    const float* bconcat = (const float*)d_in[7];
    float* out = (float*)d_out;

    // workspace layout (~1.7 MB)
    char* ws = (char*)d_ws;
    int*   idx   = (int*)ws;                         // 64 KB
    float* beff  = (float*)(ws + 65536);             // 1 KB (padded)
    float* Weff  = (float*)(ws + 66560);             // 512 KB
    float* r2arr = (float*)(ws + 590848);            // 128 KB
    float* pbest = (float*)(ws + 721920);            // 512 KB
    int*   pidx  = (int*)  (ws + 1246208);           // 512 KB

    // 1) r2 precompute: B*NC = 32768 values
    r2_kernel<<<(BB * NC) / 256, 256, 0, stream>>>(rxyz, r2arr);

    // 2) WMMA NN: 8192 waves, 8 waves/block -> 1024 blocks
    nn_wmma_kernel<<<1024, 256, 0, stream>>>(qxyz, rxyz, r2arr, pbest, pidx);

    // 3) chunk reduce: B*S threads
    nn_reduce_kernel<<<(BB * SS) / 256, 256, 0, stream>>>(pbest, pidx, idx);

    // 4) Weff = Wc2 @ Wseed : 512 wave-tiles
    weff_kernel<<<64, 256, 0, stream>>>(Wconcat, Wseed, Weff);

    // 5) beff
    beff_kernel<<<1, 256, 0, stream>>>(Wconcat, bseed, bconcat, beff);

    // 6) fused GEMM: 16384 wave-tiles -> 2048 blocks
    fused_out_kernel<<<2048, 256, 0, stream>>>(Wconcat, Weff, seed, reff,
                                               idx, beff, out);
}